// ConceptWhiteningLayer_23321672417519
// MI455X (gfx1250) — compile-verified
//
#include <hip/hip_runtime.h>
#include <hip/hip_bf16.h>
#include <math.h>

// ---------------- problem constants ----------------
#define C_DIM 256
#define HW    1024          // 32*32
#define N_IMG 64
#define M_TOT (N_IMG * HW)  // 65536
#define EPS_F 1e-5f
#define NSPLIT 16
#define KCHUNK (M_TOT / NSPLIT)  // 4096
#define KSTEP  64
#define NWG_ITER 16

typedef __attribute__((ext_vector_type(2))) float v2f;
typedef __attribute__((ext_vector_type(8))) float v8f;

__device__ __forceinline__ v8f wmma4(v2f a, v2f b, v8f c) {
  // V_WMMA_F32_16X16X4_F32 : D = A(16x4) * B(4x16) + C(16x16), fp32
  return __builtin_amdgcn_wmma_f32_16x16x4_f32(
      /*neg_a=*/false, a, /*neg_b=*/false, b,
      /*c_mod=*/(short)0, c, /*reuse_a=*/false, /*reuse_b=*/false);
}

// ---- CDNA5 async global->LDS copy (ASYNCcnt path, CDNA5 ISA) ----
// IOFFSET is added to BOTH the LDS and global addresses, so contiguous panel
// copies need only one base address pair + immediate offsets (no VALU adds).
__device__ __forceinline__ unsigned lds_addr(const void* p) {
  // flat shared address: high 32 = aperture, low 32 = LDS byte offset
  return (unsigned)(size_t)p;
}
__device__ __forceinline__ void async_b128(unsigned lds_byte, const void* gsrc, int ioff) {
  asm volatile("global_load_async_to_lds_b128 %0, %1, off offset:%2"
               :: "v"(lds_byte), "v"(gsrc), "i"(ioff) : "memory");
}
__device__ __forceinline__ void wait_async0() {
  asm volatile("s_wait_asynccnt 0x0" ::: "memory");
}

// ---------------- workspace layout (float offsets) ----------------
#define OFF_MEAN 0                       // 256
#define OFF_BVEC 256                     // 256
#define OFF_SCAL 512                     // 16: [0]=1/tr, [1]=sqrt(1/tr)
#define OFF_BAR  528                     // 16 uints (barrier state)
#define OFF_GP   1024                    // NSPLIT * 65536
#define OFF_SIG  (OFF_GP + NSPLIT * 65536)
#define OFF_P0   (OFF_SIG + 65536)
#define OFF_P1   (OFF_P0 + 65536)
#define OFF_T1   (OFF_P1 + 65536)
#define OFF_T2   (OFF_T1 + 65536)
#define OFF_W    (OFF_T2 + 65536)

// ---------------- software grid barrier (16 resident WGs) ----------------
__device__ __forceinline__ void gbar(unsigned* cnt, unsigned* gen, unsigned nwg) {
  __syncthreads();
  if (threadIdx.x == 0) {
    __threadfence();
    unsigned g = __hip_atomic_load(gen, __ATOMIC_ACQUIRE, __HIP_MEMORY_SCOPE_AGENT);
    unsigned v = __hip_atomic_fetch_add(cnt, 1u, __ATOMIC_ACQ_REL, __HIP_MEMORY_SCOPE_AGENT);
    if (v == nwg - 1u) {
      __hip_atomic_store(cnt, 0u, __ATOMIC_RELAXED, __HIP_MEMORY_SCOPE_AGENT);
      __hip_atomic_fetch_add(gen, 1u, __ATOMIC_RELEASE, __HIP_MEMORY_SCOPE_AGENT);
    } else {
      while (__hip_atomic_load(gen, __ATOMIC_ACQUIRE, __HIP_MEMORY_SCOPE_AGENT) == g)
        __builtin_amdgcn_s_sleep(2);
    }
    __threadfence();
  }
  __syncthreads();
}

// ---------------- K0: per-channel mean, barrier init ----------------
__global__ __launch_bounds__(256) void k_mean(const float* __restrict__ X,
                                              float* __restrict__ mean,
                                              unsigned* __restrict__ bar) {
  __shared__ float red[256];
  const int c = blockIdx.x;
  const int t = threadIdx.x;
  float s = 0.0f;
  for (int n = 0; n < N_IMG; ++n) {
    const float4 v = ((const float4*)(X + (((size_t)n * C_DIM + c) << 10)))[t];
    s += v.x + v.y + v.z + v.w;
  }
  red[t] = s;
  __syncthreads();
  for (int off = 128; off > 0; off >>= 1) {
    if (t < off) red[t] += red[t + off];
    __syncthreads();
  }
  if (t == 0) mean[c] = red[0] * (1.0f / (float)M_TOT);
  if (blockIdx.x == 0 && t < 16) bar[t] = 0u;
}

// ---------------- K1: split-K gram partials, fp32 WMMA ----------------
// grid = (16 tile positions, NSPLIT splits); 64x64 tile per WG, 8 waves.
// Double-buffered LDS fed by async global->LDS copies.
__global__ __launch_bounds__(256) void k_gram(const float* __restrict__ X,
                                              float* __restrict__ Gp) {
  __shared__ float As[2][64][68];
  __shared__ float Bs[2][64][68];
  const int tile  = blockIdx.x;          // 0..15
  const int split = blockIdx.y;          // 0..NSPLIT-1
  const int i0 = (tile >> 2) * 64;
  const int j0 = (tile & 3) * 64;
  const int t = threadIdx.x;
  const int wave = t >> 5, lane = t & 31;
  const int grp = lane >> 4, lm = lane & 15;
  const int sr = wave >> 1;              // 0..3
  const int sc = (wave & 1) * 2;         // 0 or 2

  const int lrow = t >> 2;               // 0..63
  const int lcol = (t & 3) * 16;         // 0/16/32/48
  const int NSTEP = KCHUNK / KSTEP;      // 64

  auto issue = [&](int buf, int step) {
    const int kbase = split * KCHUNK + step * KSTEP;
    const int n   = kbase >> 10;
    const int hw0 = kbase & 1023;
    const float* pa = X + (((size_t)n * C_DIM + (i0 + lrow)) << 10) + hw0 + lcol;
    const float* pb = X + (((size_t)n * C_DIM + (j0 + lrow)) << 10) + hw0 + lcol;
    const unsigned la = lds_addr(&As[buf][lrow][lcol]);
    const unsigned lb = lds_addr(&Bs[buf][lrow][lcol]);
#pragma unroll
    for (int q = 0; q < 4; ++q) {
      async_b128(la, pa, 16 * q);
      async_b128(lb, pb, 16 * q);
    }
  };

  v8f acc0 = {}; v8f acc1 = {};
  issue(0, 0);
  for (int step = 0; step < NSTEP; ++step) {
    const int cur = step & 1;
    wait_async0();          // this wave's async copies for `cur` are in LDS
    __syncthreads();        // ... and everyone else's too; prior reads done
    if (step + 1 < NSTEP) issue(cur ^ 1, step + 1);
#pragma unroll
    for (int k = 0; k < KSTEP; k += 4) {
      v2f a;
      a.x = As[cur][sr * 16 + lm][k + 2 * grp];
      a.y = As[cur][sr * 16 + lm][k + 2 * grp + 1];
      v2f b0, b1;
      b0.x = Bs[cur][sc * 16 + lm][k + 2 * grp];
      b0.y = Bs[cur][sc * 16 + lm][k + 2 * grp + 1];
      b1.x = Bs[cur][(sc + 1) * 16 + lm][k + 2 * grp];
      b1.y = Bs[cur][(sc + 1) * 16 + lm][k + 2 * grp + 1];
      acc0 = wmma4(a, b0, acc0);
      acc1 = wmma4(a, b1, acc1);
    }
  }
  float* out = Gp + (size_t)split * (C_DIM * C_DIM);
#pragma unroll
  for (int v = 0; v < 8; ++v) {
    const int m = v + 8 * grp;
    out[(i0 + sr * 16 + m) * C_DIM + (j0 + sc * 16 + lm)]       = acc0[v];
    out[(i0 + sr * 16 + m) * C_DIM + (j0 + (sc + 1) * 16 + lm)] = acc1[v];
  }
}

// ---------------- 256^3 matmul stripe: C = sAB*(A@B) [+ sD*D] ----------------
// WG `wg` computes rows [wg*16, wg*16+16); 8 waves x 2 col-tiles each.
__device__ void mm256(const float* __restrict__ A, const float* __restrict__ B,
                      float* __restrict__ Cm, float sAB,
                      const float* __restrict__ D, float sD, int wg, int t) {
  const int wave = t >> 5, lane = t & 31;
  const int grp = lane >> 4, lm = lane & 15;
  const int i0 = wg * 16;
  const int tc0 = wave * 2;
  v8f acc0 = {}; v8f acc1 = {};
  const float* arow = A + (size_t)(i0 + lm) * C_DIM;
#pragma unroll 4
  for (int k = 0; k < C_DIM; k += 4) {
    v2f a = *(const v2f*)&arow[k + 2 * grp];
    const float* brow = B + (size_t)(k + 2 * grp) * C_DIM;
    v2f b0, b1;
    b0.x = brow[tc0 * 16 + lm];
    b0.y = brow[C_DIM + tc0 * 16 + lm];
    b1.x = brow[(tc0 + 1) * 16 + lm];
    b1.y = brow[C_DIM + (tc0 + 1) * 16 + lm];
    acc0 = wmma4(a, b0, acc0);
    acc1 = wmma4(a, b1, acc1);
  }
#pragma unroll
  for (int v = 0; v < 8; ++v) {
    const int row = i0 + v + 8 * grp;
    const int c0 = tc0 * 16 + lm;
    const int c1 = (tc0 + 1) * 16 + lm;
    const float d0 = D ? sD * D[row * C_DIM + c0] : 0.0f;
    const float d1 = D ? sD * D[row * C_DIM + c1] : 0.0f;
    Cm[row * C_DIM + c0] = sAB * acc0[v] + d0;
    Cm[row * C_DIM + c1] = sAB * acc1[v] + d1;
  }
}

// ---------------- K2: Sigma assembly + Newton-Schulz + W,b ----------------
__global__ __launch_bounds__(256) void k_iter(const float* __restrict__ R,
                                              float* __restrict__ wf,
                                              unsigned* __restrict__ bar) {
  float* mean  = wf + OFF_MEAN;
  float* bvec  = wf + OFF_BVEC;
  float* scal  = wf + OFF_SCAL;
  float* Gp    = wf + OFF_GP;
  float* Sigma = wf + OFF_SIG;
  float* P0    = wf + OFF_P0;
  float* P1    = wf + OFF_P1;
  float* T1    = wf + OFF_T1;
  float* T2    = wf + OFF_T2;
  float* Wm    = wf + OFF_W;
  unsigned* cnt = bar + 0;
  unsigned* gen = bar + 1;

  const int wg = blockIdx.x;   // 0..15
  const int t  = threadIdx.x;
  __shared__ float red[256];

  // Sigma = sum(Gp)/m - mu*mu^T + eps*I  (4096 entries per WG)
  for (int i = 0; i < 16; ++i) {
    const int idx = wg * 4096 + i * 256 + t;
    float s = 0.0f;
    for (int sp = 0; sp < NSPLIT; ++sp) s += Gp[(size_t)sp * 65536 + idx];
    const int r = idx >> 8, c = idx & 255;
    float sig = s * (1.0f / (float)M_TOT) - mean[r] * mean[c];
    if (r == c) sig += EPS_F;
    Sigma[idx] = sig;
  }
  gbar(cnt, gen, NWG_ITER);

  // trace -> 1/tr, sqrt(1/tr)
  if (wg == 0) {
    red[t] = Sigma[t * C_DIM + t];
    __syncthreads();
    for (int off = 128; off > 0; off >>= 1) {
      if (t < off) red[t] += red[t + off];
      __syncthreads();
    }
    if (t == 0) {
      const float trr = 1.0f / red[0];
      scal[0] = trr;
      scal[1] = sqrtf(trr);
    }
  }
  gbar(cnt, gen, NWG_ITER);

  // Sigma_N (in place) and P = I
  const float trr = scal[0];
  for (int i = 0; i < 16; ++i) {
    const int idx = wg * 4096 + i * 256 + t;
    Sigma[idx] *= trr;
    const int r = idx >> 8, c = idx & 255;
    P0[idx] = (r == c) ? 1.0f : 0.0f;
  }
  gbar(cnt, gen, NWG_ITER);

  // 10 Newton-Schulz iterations: P <- 1.5P - 0.5*(P@P@P)@Sigma_N
  float* Pc = P0;
  float* Pn = P1;
  for (int it = 0; it < 10; ++it) {
    mm256(Pc, Pc, T1, 1.0f, nullptr, 0.0f, wg, t);
    gbar(cnt, gen, NWG_ITER);
    mm256(T1, Pc, T2, 1.0f, nullptr, 0.0f, wg, t);
    gbar(cnt, gen, NWG_ITER);
    mm256(T2, Sigma, Pn, -0.5f, Pc, 1.5f, wg, t);
    gbar(cnt, gen, NWG_ITER);
    float* tmp = Pc; Pc = Pn; Pn = tmp;
  }

  // W = sqrt(1/tr) * (R @ P)   (folds rotation into whitening)
  mm256(R, Pc, Wm, scal[1], nullptr, 0.0f, wg, t);
  gbar(cnt, gen, NWG_ITER);

  // b = W @ mean
  if (t < 16) {
    const int d = wg * 16 + t;
    float s = 0.0f;
    for (int c = 0; c < C_DIM; ++c) s += Wm[d * C_DIM + c] * mean[c];
    bvec[d] = s;
  }
}

// ---------------- K3: out = W @ x - b  (WMMA, 64x128 tiles) ----------------
// W stripe staged once into LDS; X panels double-buffered via async copies.
__global__ __launch_bounds__(256) void k_apply(const float* __restrict__ X,
                                               const float* __restrict__ Wm,
                                               const float* __restrict__ bvec,
                                               float* __restrict__ out) {
  __shared__ float Wsh[64][260];       // A: rows d, full 256 cols c
  __shared__ float Xsh[2][64][132];    // B: rows c (64-chunk), cols pos
  const int pc = blockIdx.x;           // 0..511 : 128-position chunk
  const int dg = blockIdx.y;           // 0..3   : 64-row group
  const int d0 = dg * 64;
  const int p0 = pc * 128;
  const int n   = p0 >> 10;
  const int hw0 = p0 & 1023;
  const int t = threadIdx.x;
  const int wave = t >> 5, lane = t & 31;
  const int grp = lane >> 4, lm = lane & 15;
  const int sr  = wave >> 1;            // 0..3
  const int sc0 = (wave & 1) * 4;       // 0 or 4 (4 col subtiles each)
  const int lrow = t >> 2;              // 0..63

  auto issueX = [&](int buf, int ci) {
    const int cb = ci * 64;
    const float* px = X + (((size_t)n * C_DIM + (cb + lrow)) << 10) + hw0 + (t & 3) * 32;
    const unsigned lx = lds_addr(&Xsh[buf][lrow][(t & 3) * 32]);
#pragma unroll
    for (int q = 0; q < 8; ++q) async_b128(lx, px, 16 * q);
  };

  // stage full W stripe (64 x 256) once, async
  {
    const float* pw = Wm + (size_t)(d0 + lrow) * C_DIM + (t & 3) * 64;
    const unsigned lw = lds_addr(&Wsh[lrow][(t & 3) * 64]);
#pragma unroll
    for (int q = 0; q < 16; ++q) async_b128(lw, pw, 16 * q);
  }
  issueX(0, 0);

  v8f acc[4] = {{}, {}, {}, {}};
  for (int ci = 0; ci < 4; ++ci) {
    const int cur = ci & 1;
    wait_async0();
    __syncthreads();
    if (ci + 1 < 4) issueX(cur ^ 1, ci + 1);
    const int cb = ci * 64;
#pragma unroll
    for (int k = 0; k < 64; k += 4) {
      v2f a;
      a.x = Wsh[sr * 16 + lm][cb + k + 2 * grp];
      a.y = Wsh[sr * 16 + lm][cb + k + 2 * grp + 1];
#pragma unroll
      for (int j = 0; j < 4; ++j) {
        v2f b;
        b.x = Xsh[cur][k + 2 * grp][(sc0 + j) * 16 + lm];
        b.y = Xsh[cur][k + 2 * grp + 1][(sc0 + j) * 16 + lm];
        acc[j] = wmma4(a, b, acc[j]);
      }
    }
  }
#pragma unroll
  for (int j = 0; j < 4; ++j) {
    const int pos = p0 + (sc0 + j) * 16 + lm;
    const int hw  = pos & 1023;
#pragma unroll
    for (int v = 0; v < 8; ++v) {
      const int d = d0 + sr * 16 + v + 8 * grp;
      out[(((size_t)n * C_DIM + d) << 10) + hw] = acc[j][v] - bvec[d];
    }
  }
}

// ---------------- host launch ----------------
extern "C" void kernel_launch(void* const* d_in, const int* in_sizes, int n_in,
                              void* d_out, int out_size, void* d_ws, size_t ws_size,
                              hipStream_t stream) {
  (void)in_sizes; (void)n_in; (void)out_size; (void)ws_size;
  const float* X = (const float*)d_in[0];         // (64,256,32,32) fp32
  const float* R = (const float*)d_in[1];         // (1,256,256)   fp32
  float* out = (float*)d_out;                     // (64,256,32,32) fp32
  float* wf  = (float*)d_ws;
  unsigned* bar = (unsigned*)(wf + OFF_BAR);

  k_mean<<<dim3(C_DIM), dim3(256), 0, stream>>>(X, wf + OFF_MEAN, bar);
  k_gram<<<dim3(16, NSPLIT), dim3(256), 0, stream>>>(X, wf + OFF_GP);
  k_iter<<<dim3(NWG_ITER), dim3(256), 0, stream>>>(R, wf, bar);
  k_apply<<<dim3(512, 4), dim3(256), 0, stream>>>(X, wf + OFF_W, wf + OFF_BVEC, out);
}